// GraphMultiheadAttention_11141145166493
// MI455X (gfx1250) — compile-verified
//
#include <hip/hip_runtime.h>
#include <math.h>

// ---------------------------------------------------------------------------
// Graph multi-head attention, MI455X (gfx1250), full f32 via V_WMMA_F32_16X16X4_F32.
// Memory-bound problem (268MB attno write dominates); f32 WMMA keeps reference
// precision at zero cost against the HBM roof.
// ---------------------------------------------------------------------------

typedef float v2f __attribute__((ext_vector_type(2)));
typedef float v8f __attribute__((ext_vector_type(8)));

constexpr int SEQ = 2048;
constexpr int NB  = 2;
constexpr int DM  = 256;
constexpr int NHH = 8;
constexpr int DKK = 32;
constexpr float NEGV = -1.0e9f;
constexpr float EPSV = 1e-6f;

static __device__ __forceinline__ v8f wmma_f32x4(v2f a, v2f b, v8f c) {
  // 8 args: (neg_a, A, neg_b, B, c_mod, C, reuse_a, reuse_b)
  return __builtin_amdgcn_wmma_f32_16x16x4_f32(false, a, false, b, (short)0, c,
                                               false, false);
}

// ---------------------------------------------------------------------------
// Kernel A: projection GEMM  y = x @ W,  x is [SEQ,NB,DM] (seq-first),
// rows are batch-first r = b*SEQ+s; output stored as [NB, NHH, SEQ, DKK].
// One wave per 16x16 output tile, K-loop of 64 x wmma_f32_16x16x4.
// ---------------------------------------------------------------------------
__global__ __launch_bounds__(32) void proj_kernel(const float* __restrict__ x,
                                                  const float* __restrict__ w,
                                                  float* __restrict__ out) {
  const int tile = blockIdx.x;
  const int mt = tile >> 4;   // 256 m-tiles over NB*SEQ rows
  const int nt = tile & 15;   // 16 n-tiles over DM cols
  const int lane = threadIdx.x & 31;
  const int m   = lane & 15;
  const int kq  = (lane >> 4) << 1;
  const int hi8 = (lane >> 4) << 3;

  const int r = mt * 16 + m;          // batch-first row index
  const int b = r >> 11;              // / SEQ
  const int s = r & (SEQ - 1);
  const float* xrow = x + ((size_t)s * NB + b) * DM;
  const int ncol = nt * 16 + m;

  v8f acc = {};
  for (int k0 = 0; k0 < DM; k0 += 4) {
    v2f a, bf;
    a.x  = xrow[k0 + kq];
    a.y  = xrow[k0 + kq + 1];
    bf.x = w[(size_t)(k0 + kq) * DM + ncol];
    bf.y = w[(size_t)(k0 + kq + 1) * DM + ncol];
    acc  = wmma_f32x4(a, bf, acc);
  }
#pragma unroll
  for (int j = 0; j < 8; ++j) {
    const int row = j + hi8;
    const int rr = mt * 16 + row;
    const int b2 = rr >> 11, s2 = rr & (SEQ - 1);
    const int n = nt * 16 + m;
    const int h = n >> 5, d = n & 31;
    out[(((size_t)b2 * NHH + h) * SEQ + s2) * DKK + d] = acc[j];
  }
}

// ---------------------------------------------------------------------------
// Kernel B: fused attention per (b, h, 16-query tile).
//   phase 1: scores = (Q/sqrt(dk)) @ K^T, masked -> LDS stripe [16 x 2048]
//   phase 2: row softmax in LDS
//   phase 3: coalesced float4 store of attno (the big 268MB output)
//   phase 4: (attno * graph_pos bias) @ V with per-wave K split + LDS reduce
// LDS: 128KB scores + 16KB partials + scratch  (< 160KB, WGP has 320KB)
// ---------------------------------------------------------------------------
__global__ __launch_bounds__(256) void attn_kernel(
    const float* __restrict__ qh, const float* __restrict__ kh,
    const float* __restrict__ vh, const float* __restrict__ gp,
    const int* __restrict__ mask, float* __restrict__ attn_out,
    float* __restrict__ ctx) {
  __shared__ __align__(16) float sc[16 * SEQ];       // 131072 B
  __shared__ __align__(16) float part[8 * 16 * 32];  // 16384 B
  __shared__ float red[16 * 16];
  __shared__ float rowstat[16];

  const int qt = blockIdx.x, h = blockIdx.y, bb = blockIdx.z;
  const int tid = threadIdx.x;
  const int wv = tid >> 5, lane = tid & 31;
  const int m   = lane & 15;
  const int kq  = (lane >> 4) << 1;
  const int hi8 = (lane >> 4) << 3;
  const float inv_temp = 0.17677669529663687f;  // 1/sqrt(32)

  // ---- phase 1: masked scaled scores into LDS -------------------------------
  const float* qbase = qh + (((size_t)bb * NHH + h) * SEQ + qt * 16 + m) * DKK;
  v2f aq[8];
#pragma unroll
  for (int kk = 0; kk < 8; ++kk) {
    aq[kk].x = qbase[kk * 4 + kq] * inv_temp;
    aq[kk].y = qbase[kk * 4 + kq + 1] * inv_temp;
  }
  const float* kbase = kh + ((size_t)bb * NHH + h) * SEQ * DKK;
  for (int kt = wv * 16; kt < wv * 16 + 16; ++kt) {
    v8f acc = {};
    const float* krow = kbase + (size_t)(kt * 16 + m) * DKK;  // B col n = key row
#pragma unroll
    for (int kk = 0; kk < 8; ++kk) {
      v2f bf;
      bf.x = krow[kk * 4 + kq];
      bf.y = krow[kk * 4 + kq + 1];
      acc  = wmma_f32x4(aq[kk], bf, acc);
    }
#pragma unroll
    for (int j = 0; j < 8; ++j) {
      const int row = j + hi8;
      const int qg = qt * 16 + row;
      const int kg = kt * 16 + m;
      const int mk = mask[((size_t)bb * SEQ + qg) * SEQ + kg];
      sc[row * SEQ + kg] = (mk == 0) ? NEGV : acc[j];
    }
  }
  __syncthreads();

  // ---- phase 2: row softmax over 2048 keys (16 threads per row) -------------
  const int r2 = tid >> 4, sub = tid & 15;
  float mx = -INFINITY;
  for (int c = sub; c < SEQ; c += 16) mx = fmaxf(mx, sc[r2 * SEQ + c]);
  red[r2 * 16 + sub] = mx;
  __syncthreads();
  if (sub == 0) {
    float v = red[r2 * 16];
    for (int i = 1; i < 16; ++i) v = fmaxf(v, red[r2 * 16 + i]);
    rowstat[r2] = v;
  }
  __syncthreads();
  const float rmax = rowstat[r2];
  float sum = 0.f;
  for (int c = sub; c < SEQ; c += 16) {
    const float e = __expf(sc[r2 * SEQ + c] - rmax);
    sc[r2 * SEQ + c] = e;
    sum += e;
  }
  red[r2 * 16 + sub] = sum;
  __syncthreads();
  if (sub == 0) {
    float v = 0.f;
    for (int i = 0; i < 16; ++i) v += red[r2 * 16 + i];
    rowstat[r2] = 1.0f / v;
  }
  __syncthreads();
  const float inv = rowstat[r2];
  for (int c = sub; c < SEQ; c += 16) sc[r2 * SEQ + c] *= inv;
  __syncthreads();

  // ---- phase 3: coalesced attno write (dominant HBM traffic) ----------------
  {
    float4* dst4 = (float4*)(attn_out +
                             (((size_t)bb * NHH + h) * SEQ + qt * 16) * SEQ);
    const float4* src4 = (const float4*)sc;
    for (int i = tid; i < 16 * SEQ / 4; i += 256) dst4[i] = src4[i];
  }

  // ---- phase 4: out = (attno * bias) @ V ; per-wave 256-key chunk -----------
  const float* vbase = vh + ((size_t)bb * NHH + h) * SEQ * DKK;
  const int qgm = qt * 16 + m;
  v8f acc0 = {}, acc1 = {};
  for (int st = 0; st < 64; ++st) {
    const int kcol = wv * 256 + st * 4 + kq;
    v2f a, b0, b1;
    // bias[h,q,k] = graph_pos[k, q, h]
    a.x = sc[m * SEQ + kcol] * gp[((size_t)kcol * SEQ + qgm) * NHH + h];
    a.y = sc[m * SEQ + kcol + 1] * gp[((size_t)(kcol + 1) * SEQ + qgm) * NHH + h];
    const float* vrow = vbase + (size_t)kcol * DKK;
    b0.x = vrow[m];            b0.y = vrow[DKK + m];
    b1.x = vrow[16 + m];       b1.y = vrow[DKK + 16 + m];
    acc0 = wmma_f32x4(a, b0, acc0);
    acc1 = wmma_f32x4(a, b1, acc1);
  }
#pragma unroll
  for (int j = 0; j < 8; ++j) {
    const int row = j + hi8;
    part[(wv * 16 + row) * 32 + m]      = acc0[j];
    part[(wv * 16 + row) * 32 + 16 + m] = acc1[j];
  }
  __syncthreads();
  for (int idx = tid; idx < 512; idx += 256) {
    const int row = idx >> 5, col = idx & 31;
    float sv = 0.f;
#pragma unroll
    for (int w2 = 0; w2 < 8; ++w2) sv += part[(w2 * 16 + row) * 32 + col];
    const int qg = qt * 16 + row;
    ctx[((size_t)bb * SEQ + qg) * DM + h * DKK + col] = sv;  // [B,S,D]
  }
}

// ---------------------------------------------------------------------------
// Kernel C: out = ctx @ w_fc + residual, LayerNorm, transpose to [SEQ,NB,DM].
// One block per 16 rows; 8 waves cover the 16 n-tiles (2 each).
// ---------------------------------------------------------------------------
__global__ __launch_bounds__(256) void fc_ln_kernel(
    const float* __restrict__ ctx, const float* __restrict__ wfc,
    const float* __restrict__ qin, const float* __restrict__ gamma,
    const float* __restrict__ beta, float* __restrict__ out) {
  __shared__ __align__(16) float y[16 * DM];
  __shared__ float redS[256];
  __shared__ float redQ[256];
  __shared__ float rowMu[16];
  __shared__ float rowRs[16];

  const int mt = blockIdx.x;
  const int tid = threadIdx.x, wv = tid >> 5, lane = tid & 31;
  const int m   = lane & 15;
  const int kq  = (lane >> 4) << 1;
  const int hi8 = (lane >> 4) << 3;
  const float* arow = ctx + (size_t)(mt * 16 + m) * DM;

  for (int t = 0; t < 2; ++t) {
    const int nt = wv * 2 + t;
    const int ncol = nt * 16 + m;
    v8f acc = {};
    for (int k0 = 0; k0 < DM; k0 += 4) {
      v2f a, b;
      a.x = arow[k0 + kq];
      a.y = arow[k0 + kq + 1];
      b.x = wfc[(size_t)(k0 + kq) * DM + ncol];
      b.y = wfc[(size_t)(k0 + kq + 1) * DM + ncol];
      acc = wmma_f32x4(a, b, acc);
    }
#pragma unroll
    for (int j = 0; j < 8; ++j) y[(j + hi8) * DM + ncol] = acc[j];
  }
  __syncthreads();

  const int r = tid >> 4, sub = tid & 15;
  const int rr = mt * 16 + r;
  const int b2 = rr >> 11, s2 = rr & (SEQ - 1);
  const float* resid = qin + ((size_t)s2 * NB + b2) * DM;  // residual = qb
  float sum = 0.f, sq = 0.f;
#pragma unroll
  for (int i = 0; i < 16; ++i) {
    const int c = sub * 16 + i;
    const float v = y[r * DM + c] + resid[c];
    y[r * DM + c] = v;
    sum += v;
    sq += v * v;
  }
  redS[r * 16 + sub] = sum;
  redQ[r * 16 + sub] = sq;
  __syncthreads();
  if (sub == 0) {
    float s = 0.f, q = 0.f;
    for (int i = 0; i < 16; ++i) { s += redS[r * 16 + i]; q += redQ[r * 16 + i]; }
    const float mu = s / DM;
    rowMu[r] = mu;
    rowRs[r] = rsqrtf(q / DM - mu * mu + EPSV);
  }
  __syncthreads();
  const float mu = rowMu[r], rs = rowRs[r];
  float* orow = out + ((size_t)s2 * NB + b2) * DM;  // back to seq-first
#pragma unroll
  for (int i = 0; i < 16; ++i) {
    const int c = sub * 16 + i;
    orow[c] = (y[r * DM + c] - mu) * rs * gamma[c] + beta[c];
  }
}

// ---------------------------------------------------------------------------
extern "C" void kernel_launch(void* const* d_in, const int* in_sizes, int n_in,
                              void* d_out, int out_size, void* d_ws,
                              size_t ws_size, hipStream_t stream) {
  (void)in_sizes; (void)n_in; (void)out_size; (void)ws_size;
  const float* q     = (const float*)d_in[0];
  const float* k     = (const float*)d_in[1];
  const float* v     = (const float*)d_in[2];
  const float* gp    = (const float*)d_in[3];
  const int*   mask  = (const int*)d_in[4];
  const float* wq    = (const float*)d_in[5];
  const float* wk    = (const float*)d_in[6];
  const float* wv    = (const float*)d_in[7];
  const float* wfc   = (const float*)d_in[8];
  const float* gamma = (const float*)d_in[9];
  const float* beta  = (const float*)d_in[10];

  float* outp = (float*)d_out;                       // [SEQ,NB,DM]
  float* attn_out = outp + (size_t)SEQ * NB * DM;    // [NB,NHH,SEQ,SEQ]

  const size_t projN = (size_t)NB * NHH * SEQ * DKK;  // 1M floats each
  float* qh  = (float*)d_ws;
  float* kh  = qh + projN;
  float* vh  = kh + projN;
  float* ctx = vh + projN;                            // [NB,SEQ,DM]

  const dim3 gProj((NB * SEQ / 16) * (DM / 16)), bProj(32);
  proj_kernel<<<gProj, bProj, 0, stream>>>(q, wq, qh);
  proj_kernel<<<gProj, bProj, 0, stream>>>(k, wk, kh);
  proj_kernel<<<gProj, bProj, 0, stream>>>(v, wv, vh);

  const dim3 gAttn(SEQ / 16, NHH, NB);
  attn_kernel<<<gAttn, dim3(256), 0, stream>>>(qh, kh, vh, gp, mask, attn_out,
                                               ctx);

  fc_ln_kernel<<<dim3(NB * SEQ / 16), dim3(256), 0, stream>>>(ctx, wfc, q,
                                                              gamma, beta,
                                                              outp);
}